// BFPQuantizer_83210696393487
// MI455X (gfx1250) — compile-verified
//
#include <hip/hip_runtime.h>
#include <stdint.h>

// Block-floating-point quantize-dequantize, BLOCK_SIZE=8 along last dim.
// x: (8192, 12284) fp32 -> y same shape.  Memory-bound: ~403 MB in + 403 MB
// out @ 23.3 TB/s ~= 35 us of raw traffic -> the kernel is purely a
// data-movement exercise.  CDNA5 path: 4-deep async global->LDS B128 ring
// (ASYNCcnt, in-order completion) + NT hints in both directions since the
// stream is 4x the 192 MB L2 and touched exactly once.

typedef float v4f __attribute__((ext_vector_type(4)));

#define ROWS  8192
#define COLS  12284
#define BS    8
#define BPR   1536                 /* blocks per row: ceil(12284/8)        */
#define TPB   256
#define NWG   3072
#define DEPTH 4                    /* async ring depth (power of two)      */

constexpr uint32_t STRIDE  = (uint32_t)NWG * TPB;        // 786,432 blocks/iter
constexpr uint32_t NBLK    = (uint32_t)ROWS * BPR;       // 12,582,912
constexpr uint32_t ITERS   = NBLK / STRIDE;              // 16 (exact)
constexpr uint32_t ROWSTEP = STRIDE / BPR;               // 512 rows/iter (exact)
static_assert(NBLK % STRIDE == 0, "uniform trip count");
static_assert(STRIDE % BPR == 0, "block-in-row is loop-invariant");
static_assert((DEPTH & (DEPTH - 1)) == 0, "ring depth power of two");

__global__ __launch_bounds__(TPB) void bfp_qdq_kernel(const float* __restrict__ x,
                                                      float* __restrict__ y) {
    __shared__ float lds[DEPTH][TPB * BS];               // 4 x 8 KB ring

    const uint32_t tid  = threadIdx.x;
    const uint32_t gtid = blockIdx.x * TPB + tid;

    // One division total: this thread's row/block decomposition is fixed;
    // each iteration just advances 512 rows.
    const uint32_t row0 = gtid / BPR;
    const uint32_t blk  = gtid - row0 * BPR;
    const bool     full = (blk * BS + BS) <= (uint32_t)COLS;  // 4-elem tail block?
    const size_t   off0 = (size_t)row0 * COLS + (size_t)blk * BS;  // 16B-aligned
    constexpr size_t STEP = (size_t)ROWSTEP * COLS;      // elements per iteration

    // Workgroup-relative LDS byte offset of this lane's 32B slot in ring buffer 0.
    const uint32_t lbase = (uint32_t)(uintptr_t)(void*)&lds[0][tid * BS];

    // Issue exactly two in-order async B128 copies (32B) into ring slot.
    auto stage = [&](const float* p0, uint32_t slot) {
        const uint32_t laddr = lbase + slot * (uint32_t)(TPB * BS * sizeof(float));
        const float*   p1    = full ? (p0 + 4) : p0;     // clamp: stay in-bounds
        asm volatile("global_load_async_to_lds_b128 %0, %1, off th:TH_LOAD_NT"
                     :: "v"(laddr), "v"(p0) : "memory");
        asm volatile("global_load_async_to_lds_b128 %0, %1, off th:TH_LOAD_NT"
                     :: "v"(laddr + 16u), "v"(p1) : "memory");
    };

    // Prologue: fill DEPTH-1 stages of the ring.
    const float* pload = x + off0;
    for (uint32_t s = 0; s < DEPTH - 1u; ++s) { stage(pload, s); pload += STEP; }

    float* pst = y + off0;
    for (uint32_t i = 0; i < ITERS; ++i) {
        const uint32_t slot = i & (DEPTH - 1u);

        if (i + (DEPTH - 1u) < ITERS) {
            stage(pload, (i + DEPTH - 1u) & (DEPTH - 1u));
            pload += STEP;
            // 3 stages (6 in-order async ops) ahead of the one we read now.
            asm volatile("s_wait_asynccnt 6" ::: "memory");
        } else {
            asm volatile("s_wait_asynccnt 0" ::: "memory");  // drain in tail
        }

        const v4f a = *(const v4f*)&lds[slot][tid * BS];
        const v4f c = *(const v4f*)&lds[slot][tid * BS + 4];

        // Shared exponent: exact match of jnp.frexp on maxabs (zero block -> e=0).
        float m = fmaxf(fmaxf(fabsf(a.x), fabsf(a.y)),
                        fmaxf(fabsf(a.z), fabsf(a.w)));
        if (full) {
            m = fmaxf(m, fmaxf(fmaxf(fabsf(c.x), fabsf(c.y)),
                               fmaxf(fabsf(c.z), fabsf(c.w))));
        }
        int e;
#if __has_builtin(__builtin_amdgcn_frexp_expf)
        e = __builtin_amdgcn_frexp_expf(m);              // v_frexp_exp_i32_f32
#else
        (void)frexpf(m, &e);
#endif
        const int up = 7 - e;   // x / step == ldexp(x, up)  (pow2 -> exact)
        const int dn = e - 7;   // q * step == ldexp(q, dn)

        auto qd = [&](float v) -> float {
            float q = rintf(ldexpf(v, up));              // v_rndne = round-half-even
            q = fminf(fmaxf(q, -128.0f), 127.0f);
            return ldexpf(q, dn);
        };

        v4f r0 = { qd(a.x), qd(a.y), qd(a.z), qd(a.w) };
        __builtin_nontemporal_store(r0, (v4f*)pst);
        if (full) {
            v4f r1 = { qd(c.x), qd(c.y), qd(c.z), qd(c.w) };
            __builtin_nontemporal_store(r1, (v4f*)(pst + 4));
        }
        pst += STEP;
    }
}

extern "C" void kernel_launch(void* const* d_in, const int* in_sizes, int n_in,
                              void* d_out, int out_size, void* d_ws, size_t ws_size,
                              hipStream_t stream) {
    const float* x = (const float*)d_in[0];
    float*       y = (float*)d_out;
    (void)in_sizes; (void)n_in; (void)out_size; (void)d_ws; (void)ws_size;
    hipLaunchKernelGGL(bfp_qdq_kernel, dim3(NWG), dim3(TPB), 0, stream, x, y);
}